// GraphMemoryBank_70660801953836
// MI455X (gfx1250) — compile-verified
//
#include <hip/hip_runtime.h>
#include <hip/hip_bf16.h>
#include <math.h>

// ---------------------------------------------------------------------------
// Problem constants (fixed by the reference)
// ---------------------------------------------------------------------------
#define NN_ 16384
#define DD_ 256
#define HH_ 4
#define BB_ 64
#define DH_ 64
#define KE_ 3
#define TK_ 5
#define NDELTA (2 * BB_ * KE_)   // 384
#define THRESH 0.3f

typedef float v2f __attribute__((ext_vector_type(2)));
typedef float v8f __attribute__((ext_vector_type(8)));
typedef int   v4i __attribute__((ext_vector_type(4)));

#define GLOBAL_AS __attribute__((address_space(1)))
#define LDS_AS    __attribute__((address_space(3)))

// gfx1250 async global->LDS path (ASYNCcnt), guarded so compile never breaks.
// Builtin prototype (clang): (v4i addrspace(1)*, v4i addrspace(3)*, Ii, Ii)
#if defined(__gfx1250__) && __has_builtin(__builtin_amdgcn_global_load_async_to_lds_b128) && __has_builtin(__builtin_amdgcn_s_wait_asynccnt)
#define USE_ASYNC_LDS 1
__device__ __forceinline__ void async_cp_b128(const float* g, float* l) {
  __builtin_amdgcn_global_load_async_to_lds_b128(
      (GLOBAL_AS v4i*)g, (LDS_AS v4i*)l, 0, 0);
}
__device__ __forceinline__ void async_wait_all() {
  __builtin_amdgcn_s_wait_asynccnt(0);
}
#else
#define USE_ASYNC_LDS 0
#endif

__device__ __forceinline__ float sigmoidf_(float x) { return 1.f / (1.f + __expf(-x)); }

// ---------------------------------------------------------------------------
// WMMA f32 16x16x4 tile cores. 256-thread block (8 wave32 waves) computes a
// 128(M) x 16(N) panel; each wave owns one 16x16 C tile.
// LDS: A tile 128x32 (row stride 36 -> 16B aligned b128 stores),
//      B tile stored K-major per column: Bs[n*36 + k].
// A frag (ISA 7.12.2): lane m(0..15) row m, VGPR0/1 = K0,K1 (lanes 16+: K2,K3)
// B frag mirrored over columns. C/D: VGPR r -> M=r / r+8, N=lane%16.
// ---------------------------------------------------------------------------
__device__ __forceinline__ v8f gemm_core_nt(
    const float* __restrict__ A, int lda, int M, int mblk,
    const float* __restrict__ W, int ldw, int nb, int K,
    float* As /*128*36*/, float* Bs /*16*36*/)
{
  const int tid  = threadIdx.x;
  const int lane = tid & 31, wave = tid >> 5;
  const int l15  = lane & 15, lh = lane >> 4;
  v8f acc = {};
  for (int kc = 0; kc < K; kc += 32) {
#if USE_ASYNC_LDS
    for (int idx = tid; idx < 1024; idx += 256) {         // 128 rows x 8 qwords
      int row = idx >> 3, ku = (idx & 7) * 4;
      int gm = mblk + row;
      if (gm < M) async_cp_b128(&A[(size_t)gm * lda + kc + ku], &As[row * 36 + ku]);
      else        *(float4*)&As[row * 36 + ku] = make_float4(0.f, 0.f, 0.f, 0.f);
    }
    if (tid < 128) {                                      // 16 cols x 8 qwords
      int n = tid >> 3, ku = (tid & 7) * 4;
      async_cp_b128(&W[(size_t)(nb + n) * ldw + kc + ku], &Bs[n * 36 + ku]);
    }
    async_wait_all();
#else
    for (int idx = tid; idx < 1024; idx += 256) {
      int row = idx >> 3, ku = (idx & 7) * 4;
      int gm = mblk + row;
      float4 v = (gm < M) ? *(const float4*)&A[(size_t)gm * lda + kc + ku]
                          : make_float4(0.f, 0.f, 0.f, 0.f);
      *(float4*)&As[row * 36 + ku] = v;
    }
    if (tid < 128) {
      int n = tid >> 3, ku = (tid & 7) * 4;
      *(float4*)&Bs[n * 36 + ku] =
          *(const float4*)&W[(size_t)(nb + n) * ldw + kc + ku];
    }
#endif
    __syncthreads();
    const float* ap = As + (wave * 16 + l15) * 36;
    const float* bp = Bs + l15 * 36;
#pragma unroll
    for (int k4 = 0; k4 < 32; k4 += 4) {
      v2f a, b;
      a.x = ap[k4 + 2 * lh];
      a.y = ap[k4 + 2 * lh + 1];
      b.x = bp[k4 + 2 * lh];
      b.y = bp[k4 + 2 * lh + 1];
      acc = __builtin_amdgcn_wmma_f32_16x16x4_f32(false, a, false, b,
                                                  (short)0, acc, false, false);
    }
    __syncthreads();
  }
  return acc;
}

// Same, but B is row-major [K][Nout] (stride ldb) -> transpose into Bs[n][k]
__device__ __forceinline__ v8f gemm_core_nn(
    const float* __restrict__ A, int lda, int M, int mblk,
    const float* __restrict__ Bg, int ldb, int nb, int K,
    float* As, float* Bs)
{
  const int tid  = threadIdx.x;
  const int lane = tid & 31, wave = tid >> 5;
  const int l15  = lane & 15, lh = lane >> 4;
  v8f acc = {};
  for (int kc = 0; kc < K; kc += 32) {
#if USE_ASYNC_LDS
    for (int idx = tid; idx < 1024; idx += 256) {
      int row = idx >> 3, ku = (idx & 7) * 4;
      int gm = mblk + row;
      if (gm < M) async_cp_b128(&A[(size_t)gm * lda + kc + ku], &As[row * 36 + ku]);
      else        *(float4*)&As[row * 36 + ku] = make_float4(0.f, 0.f, 0.f, 0.f);
    }
#else
    for (int idx = tid; idx < 1024; idx += 256) {
      int row = idx >> 3, ku = (idx & 7) * 4;
      int gm = mblk + row;
      float4 v = (gm < M) ? *(const float4*)&A[(size_t)gm * lda + kc + ku]
                          : make_float4(0.f, 0.f, 0.f, 0.f);
      *(float4*)&As[row * 36 + ku] = v;
    }
#endif
    if (tid < 128) {                                      // 32 k x 4 n-quads
      int k = tid >> 2, n4 = (tid & 3) * 4;
      float4 v = *(const float4*)&Bg[(size_t)(kc + k) * ldb + nb + n4];
      Bs[(n4 + 0) * 36 + k] = v.x;
      Bs[(n4 + 1) * 36 + k] = v.y;
      Bs[(n4 + 2) * 36 + k] = v.z;
      Bs[(n4 + 3) * 36 + k] = v.w;
    }
#if USE_ASYNC_LDS
    async_wait_all();
#endif
    __syncthreads();
    const float* ap = As + (wave * 16 + l15) * 36;
    const float* bp = Bs + l15 * 36;
#pragma unroll
    for (int k4 = 0; k4 < 32; k4 += 4) {
      v2f a, b;
      a.x = ap[k4 + 2 * lh];
      a.y = ap[k4 + 2 * lh + 1];
      b.x = bp[k4 + 2 * lh];
      b.y = bp[k4 + 2 * lh + 1];
      acc = __builtin_amdgcn_wmma_f32_16x16x4_f32(false, a, false, b,
                                                  (short)0, acc, false, false);
    }
    __syncthreads();
  }
  return acc;
}

// Generic C = scale*(A @ W^T) + bias  (W row-major [Nout][K])
__global__ __launch_bounds__(256) void k_gemm_nt(
    const float* __restrict__ A, int lda,
    const float* __restrict__ W, int ldw,
    const float* __restrict__ bias,
    float* __restrict__ C, int ldc, int M, int K, float scale)
{
  __shared__ float As[128 * 36];
  __shared__ float Bs[16 * 36];
  const int nb = blockIdx.y * 16, mblk = blockIdx.x * 128;
  v8f acc = gemm_core_nt(A, lda, M, mblk, W, ldw, nb, K, As, Bs);
  const int lane = threadIdx.x & 31, wave = threadIdx.x >> 5;
  const int l15 = lane & 15, lh = lane >> 4;
  const int n = nb + l15;
  const float bv = bias ? bias[n] : 0.f;
#pragma unroll
  for (int r = 0; r < 8; ++r) {
    int m = mblk + wave * 16 + r + 8 * lh;
    if (m < M) C[(size_t)m * ldc + n] = acc[r] * scale + bv;
  }
}

// Generic C = A @ B  (B row-major [K][Nout])
__global__ __launch_bounds__(256) void k_gemm_nn(
    const float* __restrict__ A, int lda,
    const float* __restrict__ Bg, int ldb,
    float* __restrict__ C, int ldc, int M, int K)
{
  __shared__ float As[128 * 36];
  __shared__ float Bs[16 * 36];
  const int nb = blockIdx.y * 16, mblk = blockIdx.x * 128;
  v8f acc = gemm_core_nn(A, lda, M, mblk, Bg, ldb, nb, K, As, Bs);
  const int lane = threadIdx.x & 31, wave = threadIdx.x >> 5;
  const int l15 = lane & 15, lh = lane >> 4;
  const int n = nb + l15;
#pragma unroll
  for (int r = 0; r < 8; ++r) {
    int m = mblk + wave * 16 + r + 8 * lh;
    if (m < M) C[(size_t)m * ldc + n] = acc[r];
  }
}

// Fused similarity GEMM: rows 0..63 = l2n(related_pos)·nodes_n^T (sims_w),
// rows 64..127 = l2n(query)·nodes_n^T * sigmoid(values)*recency (weighted).
__global__ __launch_bounds__(256) void k_sims(
    const float* __restrict__ qn, const float* __restrict__ nodes,
    const float* __restrict__ inv_norm, const float* __restrict__ w_node,
    float* __restrict__ C)
{
  __shared__ float As[128 * 36];
  __shared__ float Bs[16 * 36];
  const int nb = blockIdx.y * 16;
  v8f acc = gemm_core_nt(qn, DD_, 128, 0, nodes, DD_, nb, DD_, As, Bs);
  const int lane = threadIdx.x & 31, wave = threadIdx.x >> 5;
  const int l15 = lane & 15, lh = lane >> 4;
  const int n = nb + l15;
  const float cin = inv_norm[n];
  const float cw  = cin * w_node[n];
#pragma unroll
  for (int r = 0; r < 8; ++r) {
    int m = wave * 16 + r + 8 * lh;
    C[(size_t)m * NN_ + n] = acc[r] * (m >= 64 ? cw : cin);
  }
}

// ---------------------------------------------------------------------------
// Prep kernels
// ---------------------------------------------------------------------------
__global__ __launch_bounds__(256) void k_prep_nodes(
    const float* __restrict__ nodes, float* __restrict__ inv_norm)
{
  const int wave = threadIdx.x >> 5, lane = threadIdx.x & 31;
  const int node = blockIdx.x * 8 + wave;
  const float* p = nodes + (size_t)node * DD_;
  float s = 0.f;
#pragma unroll
  for (int j = 0; j < 8; ++j) { float x = p[lane + 32 * j]; s += x * x; }
#pragma unroll
  for (int o = 16; o; o >>= 1) s += __shfl_xor(s, o, 32);
  if (lane == 0) inv_norm[node] = 1.f / (sqrtf(s) + 1e-8f);
}

__global__ __launch_bounds__(256) void k_node_weight(
    const float* __restrict__ node_values, const int* __restrict__ ts,
    const int* __restrict__ total_writes, float* __restrict__ w_node)
{
  int i = blockIdx.x * 256 + threadIdx.x;
  if (i < NN_) {
    float rec = __expf(-0.01f * (float)(total_writes[0] - ts[i]));
    w_node[i] = sigmoidf_(node_values[i]) * rec;
  }
}

// rows 0..63 <- l2n(related_pos), rows 64..127 <- l2n(query)
__global__ __launch_bounds__(256) void k_prep_queries(
    const float* __restrict__ related_pos, const float* __restrict__ query,
    float* __restrict__ qn)
{
  const int wave = threadIdx.x >> 5, lane = threadIdx.x & 31;
  const int row = blockIdx.x * 8 + wave;
  const float* src = (row < 64) ? related_pos + (size_t)row * DD_
                                : query + (size_t)(row - 64) * DD_;
  float s = 0.f;
#pragma unroll
  for (int j = 0; j < 8; ++j) { float x = src[lane + 32 * j]; s += x * x; }
#pragma unroll
  for (int o = 16; o; o >>= 1) s += __shfl_xor(s, o, 32);
  float inv = 1.f / (sqrtf(s) + 1e-8f);
#pragma unroll
  for (int j = 0; j < 8; ++j)
    qn[(size_t)row * DD_ + lane + 32 * j] = src[lane + 32 * j] * inv;
}

// ---------------------------------------------------------------------------
// Small-k top-k (k<=5), one block per row, ties -> lower index (JAX top_k)
// ---------------------------------------------------------------------------
__global__ __launch_bounds__(256) void k_topk(
    const float* __restrict__ S, int ncols, int k, int* __restrict__ outIdx)
{
  const int row = blockIdx.x;
  const float* s = S + (size_t)row * ncols;
  float tv[5]; int ti[5];
#pragma unroll
  for (int j = 0; j < 5; ++j) { tv[j] = -3.4e38f; ti[j] = 0x7fffffff; }
  for (int i = threadIdx.x; i < ncols; i += 256) {
    float v = s[i];
    if (v > tv[k - 1]) {
      int p = k - 1;
      while (p > 0 && v > tv[p - 1]) { tv[p] = tv[p - 1]; ti[p] = ti[p - 1]; --p; }
      tv[p] = v; ti[p] = i;
    }
  }
  __shared__ float sv[256 * 5];
  __shared__ int   si[256 * 5];
#pragma unroll
  for (int j = 0; j < 5; ++j) {
    sv[threadIdx.x * 5 + j] = (j < k) ? tv[j] : -3.4e38f;
    si[threadIdx.x * 5 + j] = (j < k) ? ti[j] : 0x7fffffff;
  }
  __syncthreads();
  if (threadIdx.x == 0) {
    for (int slot = 0; slot < k; ++slot) {
      float bv = -3.4e38f; int bi = 0x7fffffff; int bp = 0;
      for (int c = 0; c < 256 * 5; ++c) {
        if (sv[c] > bv || (sv[c] == bv && si[c] < bi)) { bv = sv[c]; bi = si[c]; bp = c; }
      }
      outIdx[row * k + slot] = bi;
      sv[bp] = -3.4e38f; si[bp] = 0x7fffffff;
    }
  }
}

// ---------------------------------------------------------------------------
// Edge MLP -> adjacency delta list (<=384 overridden entries; adj2 never built)
// ---------------------------------------------------------------------------
__global__ __launch_bounds__(256) void k_edge_delta(
    const float* __restrict__ nodes, const float* __restrict__ adjacency,
    const int* __restrict__ nbr, const int* __restrict__ write_idx,
    const float* __restrict__ w1, const float* __restrict__ b1,
    const float* __restrict__ w2, const float* __restrict__ b2,
    const float* __restrict__ w3, const float* __restrict__ b3,
    int* __restrict__ drow, int* __restrict__ dcol, int* __restrict__ dbit)
{
  const int p = blockIdx.x;                // 0..191
  const int b = p / KE_, kk = p % KE_;
  const int w  = write_idx[b];
  const int nb = nbr[b * KE_ + kk];
  __shared__ float comb[2 * DD_];
  __shared__ float h1[DD_];
  __shared__ float h2[DD_ / 2];
  const int t = threadIdx.x;
  comb[t]        = nodes[(size_t)w  * DD_ + t];
  comb[DD_ + t]  = nodes[(size_t)nb * DD_ + t];
  __syncthreads();
  float a1 = b1[t];
  for (int i = 0; i < 2 * DD_; ++i) a1 += comb[i] * w1[(size_t)i * DD_ + t];
  h1[t] = fmaxf(a1, 0.f);
  __syncthreads();
  if (t < DD_ / 2) {
    float a2 = b2[t];
    for (int i = 0; i < DD_; ++i) a2 += h1[i] * w2[(size_t)i * (DD_ / 2) + t];
    h2[t] = fmaxf(a2, 0.f);
  }
  __syncthreads();
  if (t == 0) {
    float lg = b3[0];
    for (int i = 0; i < DD_ / 2; ++i) lg += h2[i] * w3[i];
    float strength = sigmoidf_(lg);
    float val = (nb == w) ? adjacency[(size_t)w * NN_ + nb] : strength;
    int bit = (val > THRESH) ? 1 : 0;
    drow[p] = w;  dcol[p] = nb; dbit[p] = bit;                // set #1
    drow[BB_ * KE_ + p] = nb; dcol[BB_ * KE_ + p] = w;        // symmetric set #2
    dbit[BB_ * KE_ + p] = bit;
  }
}

// ---------------------------------------------------------------------------
// Visited bitmask init + BFS hop (64 batches packed into 2 u32 per node)
// ---------------------------------------------------------------------------
__global__ void k_zero_u32(unsigned* __restrict__ p, int n) {
  int i = blockIdx.x * 256 + threadIdx.x;
  if (i < n) p[i] = 0u;
}

__global__ void k_set_seeds(const int* __restrict__ seed, unsigned* __restrict__ V) {
  int t = threadIdx.x;
  if (t < BB_ * TK_) {
    int b = t / TK_;
    int nd = seed[t];
    atomicOr(&V[2 * nd + (b >> 5)], 1u << (b & 31));
  }
}

// visited_next[j] = visited[j] | OR_{i : adj2[j][i] > 0.3} visited[i]
// adj2 symmetric -> row scan == column scan; delta overrides, last wins.
// Roofline-dominant kernel: streams 1 GB/hop -> B128 loads + prefetch.
__global__ __launch_bounds__(256) void k_bfs_hop(
    const float* __restrict__ adj,
    const unsigned* __restrict__ cur, unsigned* __restrict__ nxt,
    const int* __restrict__ drow, const int* __restrict__ dcol,
    const int* __restrict__ dbit, int ndelta)
{
  const int j = blockIdx.x;
  __shared__ int ov_col[64], ov_bit[64], ov_ord[64];
  __shared__ int ov_cnt;
  __shared__ unsigned red0, red1;
  if (threadIdx.x == 0) { ov_cnt = 0; red0 = 0u; red1 = 0u; }
  __syncthreads();
  for (int t = threadIdx.x; t < ndelta; t += 256) {
    if (drow[t] == j) {
      int p = atomicAdd(&ov_cnt, 1);
      if (p < 64) { ov_col[p] = dcol[t]; ov_bit[p] = dbit[t]; ov_ord[p] = t; }
    }
  }
  __syncthreads();
  const int cnt = (ov_cnt < 64) ? ov_cnt : 64;
  unsigned a0 = 0u, a1 = 0u;
  const float4* rowp4 = (const float4*)(adj + (size_t)j * NN_);
  const uint2*  curv  = (const uint2*)cur;
  if (cnt == 0) {
    // fast path (>97% of rows): pure stream
    for (int q = threadIdx.x; q < NN_ / 4; q += 256) {
      __builtin_prefetch(rowp4 + q + 512, 0, 1);   // global_prefetch ahead
      float4 v = rowp4[q];
      int base = q * 4;
      if (v.x > THRESH) { uint2 c = curv[base + 0]; a0 |= c.x; a1 |= c.y; }
      if (v.y > THRESH) { uint2 c = curv[base + 1]; a0 |= c.x; a1 |= c.y; }
      if (v.z > THRESH) { uint2 c = curv[base + 2]; a0 |= c.x; a1 |= c.y; }
      if (v.w > THRESH) { uint2 c = curv[base + 3]; a0 |= c.x; a1 |= c.y; }
    }
  } else {
    for (int q = threadIdx.x; q < NN_ / 4; q += 256) {
      float4 v = rowp4[q];
      int base = q * 4;
      bool e0 = v.x > THRESH, e1 = v.y > THRESH, e2 = v.z > THRESH, e3 = v.w > THRESH;
      int bo0 = -1, bo1 = -1, bo2 = -1, bo3 = -1;
      for (int t = 0; t < cnt; ++t) {
        int c = ov_col[t] - base;
        if ((unsigned)c < 4u) {
          bool bit = ov_bit[t] != 0;
          int o = ov_ord[t];
          if (c == 0 && o > bo0) { bo0 = o; e0 = bit; }
          else if (c == 1 && o > bo1) { bo1 = o; e1 = bit; }
          else if (c == 2 && o > bo2) { bo2 = o; e2 = bit; }
          else if (c == 3 && o > bo3) { bo3 = o; e3 = bit; }
        }
      }
      if (e0) { uint2 c = curv[base + 0]; a0 |= c.x; a1 |= c.y; }
      if (e1) { uint2 c = curv[base + 1]; a0 |= c.x; a1 |= c.y; }
      if (e2) { uint2 c = curv[base + 2]; a0 |= c.x; a1 |= c.y; }
      if (e3) { uint2 c = curv[base + 3]; a0 |= c.x; a1 |= c.y; }
    }
  }
  if (a0) atomicOr(&red0, a0);
  if (a1) atomicOr(&red1, a1);
  __syncthreads();
  if (threadIdx.x == 0) {
    nxt[2 * j]     = cur[2 * j]     | red0;
    nxt[2 * j + 1] = cur[2 * j + 1] | red1;
  }
}

// ---------------------------------------------------------------------------
// Masked softmax over scores[h][b][:] in place (masked -> -1e9, exp underflow)
// ---------------------------------------------------------------------------
__global__ __launch_bounds__(256) void k_softmax_mask(
    float* __restrict__ S, const unsigned* __restrict__ vis)
{
  const int row = blockIdx.x;          // h*64 + b
  const int b = row & 63;
  float* s = S + (size_t)row * NN_;
  const int wsel = (b >> 5) & 1;
  const unsigned bit = 1u << (b & 31);
  __shared__ float red[256];
  float mx = -3.4e38f;
  for (int i = threadIdx.x; i < NN_; i += 256) {
    float x = (vis[2 * i + wsel] & bit) ? s[i] : -1e9f;
    mx = fmaxf(mx, x);
  }
  red[threadIdx.x] = mx; __syncthreads();
  for (int o = 128; o; o >>= 1) { if (threadIdx.x < o) red[threadIdx.x] = fmaxf(red[threadIdx.x], red[threadIdx.x + o]); __syncthreads(); }
  mx = red[0]; __syncthreads();
  float sum = 0.f;
  for (int i = threadIdx.x; i < NN_; i += 256) {
    float x = (vis[2 * i + wsel] & bit) ? s[i] : -1e9f;
    sum += __expf(x - mx);
  }
  red[threadIdx.x] = sum; __syncthreads();
  for (int o = 128; o; o >>= 1) { if (threadIdx.x < o) red[threadIdx.x] += red[threadIdx.x + o]; __syncthreads(); }
  const float inv = 1.f / red[0];
  for (int i = threadIdx.x; i < NN_; i += 256) {
    float x = (vis[2 * i + wsel] & bit) ? s[i] : -1e9f;
    s[i] = __expf(x - mx) * inv;
  }
}

// ---------------------------------------------------------------------------
// GRU combine: evolved = (1-z)*tanh(i_n + r*h_n) + z*attended
// ---------------------------------------------------------------------------
__global__ __launch_bounds__(256) void k_gru(
    const float* __restrict__ gi, const float* __restrict__ gh,
    const float* __restrict__ att, float* __restrict__ out)
{
  const int idx = blockIdx.x * 256 + threadIdx.x;   // 0..16383
  const int b = idx >> 8, o = idx & 255;
  const float ir = gi[b * 768 + o],        hr = gh[b * 768 + o];
  const float iz = gi[b * 768 + 256 + o],  hz = gh[b * 768 + 256 + o];
  const float in_ = gi[b * 768 + 512 + o], hn = gh[b * 768 + 512 + o];
  const float r = sigmoidf_(ir + hr);
  const float z = sigmoidf_(iz + hz);
  const float nn = tanhf(in_ + r * hn);
  out[idx] = (1.f - z) * nn + z * att[idx];
}

// ---------------------------------------------------------------------------
// Host orchestration
// ---------------------------------------------------------------------------
extern "C" void kernel_launch(void* const* d_in, const int* in_sizes, int n_in,
                              void* d_out, int out_size, void* d_ws, size_t ws_size,
                              hipStream_t stream)
{
  (void)in_sizes; (void)n_in; (void)out_size; (void)ws_size;
  const float* nodes       = (const float*)d_in[0];
  const float* node_values = (const float*)d_in[1];
  const float* adjacency   = (const float*)d_in[2];
  const float* related_pos = (const float*)d_in[3];
  const float* query       = (const float*)d_in[4];
  const float* w_in        = (const float*)d_in[5];
  const float* b_in        = (const float*)d_in[6];
  const float* w_out       = (const float*)d_in[7];
  const float* b_out       = (const float*)d_in[8];
  const float* rel_w1      = (const float*)d_in[9];
  const float* rel_b1      = (const float*)d_in[10];
  const float* rel_w2      = (const float*)d_in[11];
  const float* rel_b2      = (const float*)d_in[12];
  const float* rel_w3      = (const float*)d_in[13];
  const float* rel_b3      = (const float*)d_in[14];
  const float* gru_w_ih    = (const float*)d_in[15];
  const float* gru_w_hh    = (const float*)d_in[16];
  const float* gru_b_ih    = (const float*)d_in[17];
  const float* gru_b_hh    = (const float*)d_in[18];
  const int*   node_ts     = (const int*)d_in[19];
  const int*   write_idx   = (const int*)d_in[20];
  const int*   total_writes= (const int*)d_in[21];
  float* out = (float*)d_out;

  char* ws = (char*)d_ws;
  size_t off = 0;
  auto carve = [&](size_t bytes) -> void* {
    off = (off + 255) & ~(size_t)255;
    void* p = ws + off; off += bytes; return p;
  };
  float*    inv_norm = (float*)carve((size_t)NN_ * 4);
  float*    w_node   = (float*)carve((size_t)NN_ * 4);
  float*    qn       = (float*)carve((size_t)128 * DD_ * 4);
  float*    simsbuf  = (float*)carve((size_t)128 * NN_ * 4);       // 8 MB
  int*      nbr      = (int*)carve((size_t)BB_ * KE_ * 4);
  int*      seed     = (int*)carve((size_t)BB_ * TK_ * 4);
  int*      drow     = (int*)carve((size_t)NDELTA * 4);
  int*      dcol     = (int*)carve((size_t)NDELTA * 4);
  int*      dbit     = (int*)carve((size_t)NDELTA * 4);
  unsigned* V0       = (unsigned*)carve((size_t)NN_ * 2 * 4);
  unsigned* V1       = (unsigned*)carve((size_t)NN_ * 2 * 4);
  unsigned* V2       = (unsigned*)carve((size_t)NN_ * 2 * 4);
  float*    qbuf     = (float*)carve((size_t)BB_ * DD_ * 4);
  float*    kv       = (float*)carve((size_t)NN_ * 512 * 4);       // 32 MB (K|V)
  float*    scores   = (float*)carve((size_t)HH_ * BB_ * NN_ * 4); // 16 MB
  float*    ctx      = (float*)carve((size_t)BB_ * DD_ * 4);
  float*    att      = (float*)carve((size_t)BB_ * DD_ * 4);
  float*    gi       = (float*)carve((size_t)BB_ * 768 * 4);
  float*    gh       = (float*)carve((size_t)BB_ * 768 * 4);

  // 1) per-node inv-norm + weight, per-query normalization
  k_prep_nodes  <<<NN_ / 8, 256, 0, stream>>>(nodes, inv_norm);
  k_node_weight <<<NN_ / 256, 256, 0, stream>>>(node_values, node_ts, total_writes, w_node);
  k_prep_queries<<<16, 256, 0, stream>>>(related_pos, query, qn);

  // 2) fused similarity GEMM (WMMA): rows 0-63 sims_w, 64-127 weighted
  k_sims<<<dim3(1, NN_ / 16), 256, 0, stream>>>(qn, nodes, inv_norm, w_node, simsbuf);

  // 3) top-k: 3 edge neighbors (from sims_w), 5 seeds (from weighted)
  k_topk<<<BB_, 256, 0, stream>>>(simsbuf, NN_, KE_, nbr);
  k_topk<<<BB_, 256, 0, stream>>>(simsbuf + (size_t)64 * NN_, NN_, TK_, seed);

  // 4) edge MLP -> delta list (adj2 never materialized)
  k_edge_delta<<<BB_ * KE_, 256, 0, stream>>>(nodes, adjacency, nbr, write_idx,
                                              rel_w1, rel_b1, rel_w2, rel_b2,
                                              rel_w3, rel_b3, drow, dcol, dbit);

  // 5) seed visited bitmask, 2 BFS hops over adjacency (+deltas)
  k_zero_u32 <<<(NN_ * 2 + 255) / 256, 256, 0, stream>>>(V0, NN_ * 2);
  k_set_seeds<<<1, 320, 0, stream>>>(seed, V0);
  k_bfs_hop<<<NN_, 256, 0, stream>>>(adjacency, V0, V1, drow, dcol, dbit, NDELTA);
  k_bfs_hop<<<NN_, 256, 0, stream>>>(adjacency, V1, V2, drow, dcol, dbit, NDELTA);

  // 6) projections (WMMA): Q (64x256), K|V (16384x512)
  k_gemm_nt<<<dim3(1, DD_ / 16), 256, 0, stream>>>(query, DD_, w_in, DD_, b_in,
                                                   qbuf, DD_, BB_, DD_, 1.f);
  k_gemm_nt<<<dim3(NN_ / 128, 512 / 16), 256, 0, stream>>>(
      nodes, DD_, w_in + 256 * DD_, DD_, b_in + 256, kv, 512, NN_, DD_, 1.f);

  // 7) per-head scores (scale 1/sqrt(64)), masked softmax, ctx = attn @ V
  for (int h = 0; h < HH_; ++h) {
    k_gemm_nt<<<dim3(1, NN_ / 16), 256, 0, stream>>>(
        qbuf + h * DH_, DD_, kv + h * DH_, 512, nullptr,
        scores + (size_t)h * BB_ * NN_, NN_, BB_, DH_, 0.125f);
  }
  k_softmax_mask<<<HH_ * BB_, 256, 0, stream>>>(scores, V2);
  for (int h = 0; h < HH_; ++h) {
    k_gemm_nn<<<dim3(1, DH_ / 16), 256, 0, stream>>>(
        scores + (size_t)h * BB_ * NN_, NN_, kv + 256 + h * DH_, 512,
        ctx + h * DH_, DD_, BB_, NN_);
  }

  // 8) output projection + GRU
  k_gemm_nt<<<dim3(1, DD_ / 16), 256, 0, stream>>>(ctx, DD_, w_out, DD_, b_out,
                                                   att, DD_, BB_, DD_, 1.f);
  k_gemm_nt<<<dim3(1, 768 / 16), 256, 0, stream>>>(query, DD_, gru_w_ih, DD_, gru_b_ih,
                                                   gi, 768, BB_, DD_, 1.f);
  k_gemm_nt<<<dim3(1, 768 / 16), 256, 0, stream>>>(att, DD_, gru_w_hh, DD_, gru_b_hh,
                                                   gh, 768, BB_, DD_, 1.f);
  k_gru<<<BB_, 256, 0, stream>>>(gi, gh, att, out);
}